// MeanSpanRepr_65068754535064
// MI455X (gfx1250) — compile-verified
//
#include <hip/hip_runtime.h>
#include <hip/hip_bf16.h>

typedef __attribute__((ext_vector_type(2))) float v2f;
typedef __attribute__((ext_vector_type(8))) float v8f;

#define NB 32       // batches
#define TT 512      // sequence length
#define DD 256      // feature dim
#define MAX_GROUPS 160  // sum_b ceil(count_b/16) <= Q/16 + NB = 128 + 32

// -------- Phase A: counting-sort queries by batch, build 16-row groups -----
__global__ void span_bucket_kernel(const int* __restrict__ qb, int Q,
                                   int* __restrict__ counts,
                                   int* __restrict__ cursor,
                                   int* __restrict__ g_batch,
                                   int* __restrict__ g_rowstart,
                                   int* __restrict__ g_rowcount,
                                   int* __restrict__ numGroups,
                                   int* __restrict__ order) {
  int tid = threadIdx.x;
  if (tid < NB) counts[tid] = 0;
  __syncthreads();
  for (int q = tid; q < Q; q += blockDim.x)
    atomicAdd(&counts[qb[q]], 1);
  __syncthreads();
  if (tid == 0) {
    int acc = 0, ng = 0;
    for (int b = 0; b < NB; ++b) {
      cursor[b] = acc;
      int cnt = counts[b];
      for (int r = 0; r < cnt; r += 16) {
        g_batch[ng] = b;
        g_rowstart[ng] = acc + r;
        g_rowcount[ng] = (cnt - r) < 16 ? (cnt - r) : 16;
        ++ng;
      }
      acc += cnt;
    }
    *numGroups = ng;
  }
  __syncthreads();
  for (int q = tid; q < Q; q += blockDim.x) {
    int pos = atomicAdd(&cursor[qb[q]], 1);
    order[pos] = q;
  }
}

// -------- Phase B: per group (16 queries, same batch) x 16-wide d-tile, ----
// -------- fp32 WMMA: C[16x16] += A(mask,16x4) * B(X,4x16) over t-chunks ----
__global__ void __launch_bounds__(256)
span_wmma_kernel(const float* __restrict__ enc,
                 const int* __restrict__ order,
                 const int* __restrict__ g_batch,
                 const int* __restrict__ g_rowstart,
                 const int* __restrict__ g_rowcount,
                 const int* __restrict__ numGroups,
                 const int* __restrict__ start_ids,
                 const int* __restrict__ end_ids,
                 float* __restrict__ out) {
  __shared__ int sh_q[16], sh_s[16], sh_e[16];
  __shared__ float sh_inv[16];
  __shared__ int sh_tmin, sh_tmax;

  int g = blockIdx.x;
  if (g >= *numGroups) return;          // block-uniform exit

  int batch = g_batch[g];
  int rowstart = g_rowstart[g];
  int nrows = g_rowcount[g];
  int tid = threadIdx.x;

  if (tid < 16) {
    if (tid < nrows) {
      int q = order[rowstart + tid];
      int s = start_ids[q], e = end_ids[q];
      sh_q[tid] = q;
      sh_s[tid] = s;
      sh_e[tid] = e;
      sh_inv[tid] = 1.0f / (float)(e - s + 1);  // one divide per row per block
    } else {
      sh_q[tid] = -1;
      sh_s[tid] = 1;                    // empty span -> all-zero mask row
      sh_e[tid] = 0;
      sh_inv[tid] = 0.0f;
    }
  }
  __syncthreads();
  if (tid == 0) {
    int tmn = TT - 1, tmx = 0;
    for (int r = 0; r < 16; ++r) {
      if (sh_q[r] >= 0) {
        tmn = sh_s[r] < tmn ? sh_s[r] : tmn;
        tmx = sh_e[r] > tmx ? sh_e[r] : tmx;
      }
    }
    sh_tmin = tmn;
    sh_tmax = tmx;
  }
  __syncthreads();
  int tmin = sh_tmin, tmax = sh_tmax;

  int wave = tid >> 5;
  int lane = tid & 31;
  int dtile = wave + 8 * blockIdx.y;    // 0..15 -> covers D=256
  int d0 = dtile * 16;
  int mn = lane & 15;                   // A row m / B,C col n
  int kb = (lane >> 4) << 1;            // K sub-block base: 0 or 2

  int sA = sh_s[mn], eA = sh_e[mn];     // mask bounds for A row m = mn
  const float* Xb = enc + (size_t)batch * TT * DD + d0 + mn;  // col n = mn

  v8f c = {};
  int tbase = tmin & ~3;                // absolute 4-aligned chunk grid =>
                                        // bitwise deterministic accumulation
  int t0 = tbase;
  // Lane-base pointer at t = tbase + kb; advances by 16 rows per main iter.
  const float* P = Xb + (size_t)(tbase + kb) * DD;

  // Main loop: K=16 per trip = 4 WMMAs, 8 loads w/ constant immediate offsets.
  // No clamping needed: max t touched = t0 + 15 <= tmax <= TT-1.
  for (; t0 + 15 <= tmax; t0 += 16, P += 16 * DD) {
    __builtin_prefetch(P + 16 * DD, 0, 1);         // global_prefetch_b8
#pragma unroll
    for (int ch = 0; ch < 4; ++ch) {
      int ta = t0 + 4 * ch + kb;
      v2f a, bv;
      a.x = (ta >= sA && ta <= eA) ? 1.0f : 0.0f;
      a.y = (ta + 1 >= sA && ta + 1 <= eA) ? 1.0f : 0.0f;
      bv.x = P[ch * 4 * DD];                       // immediate offsets
      bv.y = P[ch * 4 * DD + DD];
      c = __builtin_amdgcn_wmma_f32_16x16x4_f32(
          false, a, false, bv, (short)0, c, false, false);
    }
  }

  // Tail: <=4 chunks, clamped loads (mask is exactly 0 beyond tmax).
  for (; t0 <= tmax; t0 += 4) {
    int ta = t0 + kb;
    int tb = ta + 1;
    v2f a, bv;
    a.x = (ta >= sA && ta <= eA) ? 1.0f : 0.0f;
    a.y = (tb >= sA && tb <= eA) ? 1.0f : 0.0f;
    int tca = ta < (TT - 1) ? ta : (TT - 1);
    int tcb = tb < (TT - 1) ? tb : (TT - 1);
    bv.x = Xb[(size_t)tca * DD];
    bv.y = Xb[(size_t)tcb * DD];
    c = __builtin_amdgcn_wmma_f32_16x16x4_f32(
        false, a, false, bv, (short)0, c, false, false);
  }

  // C layout: vgpr r, lanes 0-15 -> M=r, lanes 16-31 -> M=r+8; N = lane&15
  int hi = (lane >> 4) * 8;
#pragma unroll
  for (int r = 0; r < 8; ++r) {
    int m = r + hi;
    int q = sh_q[m];
    if (q >= 0) {
      out[(size_t)q * DD + d0 + mn] = c[r] * sh_inv[m];
    }
  }
}

extern "C" void kernel_launch(void* const* d_in, const int* in_sizes, int n_in,
                              void* d_out, int out_size, void* d_ws, size_t ws_size,
                              hipStream_t stream) {
  const float* enc = (const float*)d_in[0];
  const int* start_ids = (const int*)d_in[1];
  const int* end_ids = (const int*)d_in[2];
  const int* qb = (const int*)d_in[3];
  float* out = (float*)d_out;
  int Q = in_sizes[1];

  // workspace layout (int words)
  int* w = (int*)d_ws;
  int* counts     = w + 0;                 // 32
  int* cursor     = w + 64;                // 32
  int* g_batch    = w + 128;               // MAX_GROUPS
  int* g_rowstart = w + 128 + MAX_GROUPS;  // MAX_GROUPS
  int* g_rowcount = w + 128 + 2 * MAX_GROUPS;
  int* numGroups  = w + 128 + 3 * MAX_GROUPS;
  int* order      = w + 128 + 3 * MAX_GROUPS + 32;  // Q

  span_bucket_kernel<<<1, 256, 0, stream>>>(qb, Q, counts, cursor, g_batch,
                                            g_rowstart, g_rowcount, numGroups,
                                            order);
  span_wmma_kernel<<<dim3(MAX_GROUPS, 2), 256, 0, stream>>>(
      enc, order, g_batch, g_rowstart, g_rowcount, numGroups, start_ids,
      end_ids, out);
}